// DGLGNN_node_21423296872970
// MI455X (gfx1250) — compile-verified
//
#include <hip/hip_runtime.h>

typedef __attribute__((ext_vector_type(2))) float v2f;
typedef __attribute__((ext_vector_type(8))) float v8f;
typedef unsigned int u32x4 __attribute__((ext_vector_type(4)));
typedef int i32x4 __attribute__((ext_vector_type(4)));
typedef int i32x8 __attribute__((ext_vector_type(8)));

#if defined(__has_builtin)
#if __has_builtin(__builtin_amdgcn_tensor_load_to_lds) && \
    __has_builtin(__builtin_amdgcn_s_wait_tensorcnt)
#define USE_TDM 1
#if defined(__has_include)
#if __has_include(<hip/amd_detail/amd_gfx1250_TDM.h>)
#define TDM_ARITY6 1
#endif
#endif
#endif
#endif

__device__ __forceinline__ float lrelu(float v) { return v >= 0.f ? v : 0.1f * v; }

// ---------------------------------------------------------------- degrees
__global__ void zero_u32_kernel(unsigned* __restrict__ p, int n) {
  int i = blockIdx.x * blockDim.x + threadIdx.x;
  if (i < n) p[i] = 0u;
}

__global__ void edge_deg_kernel(const int* __restrict__ src, const int* __restrict__ dst,
                                unsigned* __restrict__ din, unsigned* __restrict__ dout, int E) {
  int e = blockIdx.x * blockDim.x + threadIdx.x;
  if (e < E) {
    atomicAdd(&din[dst[e]], 1u);
    atomicAdd(&dout[src[e]], 1u);
  }
}

__global__ void deg_fin_kernel(const unsigned* __restrict__ din, const unsigned* __restrict__ dout,
                               float* __restrict__ dinv_in, float* __restrict__ rdeg_in,
                               float* __restrict__ dinv_out, float* __restrict__ rdeg_out, int n) {
  int i = blockIdx.x * blockDim.x + threadIdx.x;
  if (i < n) {
    float a = (float)din[i] + 1.0f;
    dinv_in[i] = rsqrtf(a);
    rdeg_in[i] = 1.0f / a;
    float b = (float)dout[i] + 1.0f;
    dinv_out[i] = rsqrtf(b);
    rdeg_out[i] = 1.0f / b;
  }
}

// ---------------------------------------------------------------- encoders stage 1
__global__ __launch_bounds__(256) void enc1_inst_kernel(const float* __restrict__ x,
                                                        const float* __restrict__ W1,
                                                        const float* __restrict__ b1,
                                                        float* __restrict__ h1, int n_inst) {
  __shared__ float xs[11];
  int node = blockIdx.x;
  if (node >= n_inst) return;
  int t = threadIdx.x;
  if (t < 11) xs[t] = x[node * 11 + t];
  __syncthreads();
  float acc = b1[t];
#pragma unroll
  for (int k = 0; k < 11; k++) acc += W1[t * 11 + k] * xs[k];
  h1[(long)node * 256 + t] = lrelu(acc);
}

__global__ __launch_bounds__(128) void enc1_net_kernel(const float* __restrict__ x,
                                                       const float* __restrict__ W1,
                                                       const float* __restrict__ b1,
                                                       float* __restrict__ h1, int n_net) {
  __shared__ float xs[3];
  int node = blockIdx.x;
  if (node >= n_net) return;
  int t = threadIdx.x;
  if (t < 3) xs[t] = x[node * 3 + t];
  __syncthreads();
  float acc = b1[t];
#pragma unroll
  for (int k = 0; k < 3; k++) acc += W1[t * 3 + k] * xs[k];
  h1[(long)node * 128 + t] = lrelu(acc);
}

// ---------------------------------------------------------------- WMMA GEMM
// x[m][n] = A[m][:K] . W[n][:K] + bias[n]  (x = A @ W^T + b), Ncols = 128 fixed.
// mode 0: out = lrelu(v)
// mode 1: msg = dinv[m]*max(v,0); acc  = max(v+root[n],0)*rdeg[m]
// mode 2: msg = dinv[m]*max(v,0); acc += max(v+root[n],0)*rdeg[m]
// 256 threads = 8 waves; each wave owns a 32-row x 128-col strip (2 M-strips of 16),
// so each B fragment from LDS feeds two V_WMMA_F32_16X16X4_F32.
// W is staged into LDS as [128][K+2] (pad 2 dwords/row) — via the Tensor Data Mover
// when available (1-row tensor, pad_interval=K dwords, pad_amount=2 dwords).
__global__ __launch_bounds__(256) void wmma_gemm_kernel(
    const float* __restrict__ A, const float* __restrict__ W, const float* __restrict__ bias,
    int M, int K, int mode, float* __restrict__ out, float* __restrict__ msg,
    float* __restrict__ acc, const float* __restrict__ root, const float* __restrict__ dinv,
    const float* __restrict__ rdeg) {
  extern __shared__ float sW[];  // [128][K+2]
  const int Kp = K + 2;
  const int tid = threadIdx.x;

#if USE_TDM
  if (tid < 32) {  // wave 0 issues one TDM descriptor; TENSORcnt is per-wave
    const unsigned long long ga = (unsigned long long)W;
    const unsigned total = 128u * (unsigned)K;       // dwords to move
    const unsigned padiv = (K == 256) ? 7u : 6u;     // pad_interval: 2<<7=256, 2<<6=128 dwords
    u32x4 g0;
    g0.x = 1u;                                       // count=1 (valid descriptor)
    g0.y = 0u;                                       // lds_addr: dynamic LDS base (no static LDS)
    g0.z = (unsigned)(ga & 0xFFFFFFFFull);           // global_addr[31:0]
    g0.w = (unsigned)((ga >> 32) & 0x1FFFFFFull) | (2u << 30);  // global_addr[56:32], type=2
    i32x8 g1;
    g1[0] = (int)((2u << 16) |                       // data_size = 4B
                  (1u << 20) |                       // pad_enable
                  (padiv << 22) |                    // pad_interval = K dwords
                  (1u << 25));                       // pad_amount = 2 dwords
    g1[1] = (int)((total & 0xFFFFu) << 16);          // tensor_dim0[15:0]
    g1[2] = (int)((total >> 16) | (1u << 16));       // tensor_dim0[31:16], tensor_dim1 = 1
    g1[3] = (int)((total & 0xFFFFu) << 16);          // tile_dim0 = total dwords
    g1[4] = 0;                                       // tile_dim1 = 0 (unused), tile_dim2 = 0
    g1[5] = (int)total;                              // tensor_dim0_stride[31:0]
    g1[6] = 0;
    g1[7] = 0;
    i32x4 gz = {0, 0, 0, 0};
#if TDM_ARITY6
    i32x8 gz8 = {0, 0, 0, 0, 0, 0, 0, 0};
    __builtin_amdgcn_tensor_load_to_lds(g0, g1, gz, gz, gz8, 0);
#else
    __builtin_amdgcn_tensor_load_to_lds(g0, g1, gz, gz, 0);
#endif
    __builtin_amdgcn_s_wait_tensorcnt(0);
  }
#else
  {
    const int total4 = (128 * K) >> 2;
    for (int i4 = tid; i4 < total4; i4 += 256) {
      int i = i4 << 2;
      int n = i / K;
      int k = i - n * K;
      const float4 w4 = *(const float4*)(W + i);
      float* p = &sW[n * Kp + k];
      p[0] = w4.x; p[1] = w4.y; p[2] = w4.z; p[3] = w4.w;
    }
  }
#endif
  __syncthreads();

  const int wave = tid >> 5;
  const int lane = tid & 31;
  const int mlo = lane & 15;
  const int hi = lane >> 4;  // K pair select within each k0 step (A/B VGPR layout)
  const long rb0 = (long)blockIdx.x * 256 + wave * 32;
  long m0 = rb0 + mlo;
  long m1 = rb0 + 16 + mlo;
  // clamp instead of predicating: OOB A rows only pollute OOB C rows (never stored)
  const long Mm1 = (long)M - 1;
  const float* arow0 = A + (m0 < (long)M ? m0 : Mm1) * (long)K;
  const float* arow1 = A + (m1 < (long)M ? m1 : Mm1) * (long)K;

  v8f c0[8] = {};
  v8f c1[8] = {};

  for (int k0 = 0; k0 < K; k0 += 4) {
    const int ka = k0 + 2 * hi;
    const v2f a0 = *(const v2f*)(arow0 + ka);
    const v2f a1 = *(const v2f*)(arow1 + ka);
    v2f b[8];
#pragma unroll
    for (int t = 0; t < 8; t++) b[t] = *(const v2f*)(&sW[(t * 16 + mlo) * Kp + ka]);
#pragma unroll
    for (int t = 0; t < 8; t++) {
      c0[t] = __builtin_amdgcn_wmma_f32_16x16x4_f32(false, a0, false, b[t], (short)0, c0[t],
                                                    false, false);
      c1[t] = __builtin_amdgcn_wmma_f32_16x16x4_f32(false, a1, false, b[t], (short)0, c1[t],
                                                    false, false);
    }
  }

  // epilogue
  float bn[8], rn[8];
#pragma unroll
  for (int t = 0; t < 8; t++) {
    int n = t * 16 + mlo;
    bn[t] = bias[n];
    rn[t] = (mode >= 1) ? root[n] : 0.f;
  }
#pragma unroll
  for (int s = 0; s < 2; s++) {
    const v8f* cc = s ? c1 : c0;
    const long rbase = rb0 + s * 16;
#pragma unroll
    for (int r = 0; r < 8; r++) {
      long mm = rbase + r + 8 * hi;  // C/D layout: VGPR r = rows r (lanes 0-15), r+8 (16-31)
      if (mm >= (long)M) continue;
      float dm = 0.f, rd = 0.f;
      if (mode >= 1) {
        dm = dinv[mm];
        rd = rdeg[mm];
      }
#pragma unroll
      for (int t = 0; t < 8; t++) {
        int n = t * 16 + mlo;
        float v = cc[t][r] + bn[t];
        long o = mm * 128 + n;
        if (mode == 0) {
          out[o] = lrelu(v);
        } else {
          msg[o] = dm * fmaxf(v, 0.f);
          float sl = fmaxf(v + rn[t], 0.f) * rd;
          if (mode == 1) acc[o] = sl;
          else           acc[o] += sl;
        }
      }
    }
  }
}

// ---------------------------------------------------------------- edge scatter
// acc[to[e]][:] += dinv_to[to[e]] * msg[from[e]][:]
__global__ __launch_bounds__(256) void scatter_kernel(const int* __restrict__ efrom,
                                                      const int* __restrict__ eto,
                                                      const float* __restrict__ dinv_to,
                                                      const float* __restrict__ msg,
                                                      float* __restrict__ acc, int E) {
  int idx = blockIdx.x * blockDim.x + threadIdx.x;
  int e = idx >> 5;
  if (e >= E) return;
  int c = (idx & 31) << 2;
  int f = efrom[e];
  int t = eto[e];
  float w = dinv_to[t];
  const float4 m4 = *(const float4*)(msg + (long)f * 128 + c);
  float* ap = acc + (long)t * 128 + c;
  unsafeAtomicAdd(ap + 0, w * m4.x);
  unsafeAtomicAdd(ap + 1, w * m4.y);
  unsafeAtomicAdd(ap + 2, w * m4.z);
  unsafeAtomicAdd(ap + 3, w * m4.w);
}

// ---------------------------------------------------------------- output col-block 0
__global__ void copy_net_h0_kernel(const float* __restrict__ h, float* __restrict__ out,
                                   int n_net, int n_inst, int outw) {
  int i = blockIdx.x * blockDim.x + threadIdx.x;
  if (i < n_net * 128) {
    int node = i >> 7;
    int d = i & 127;
    out[(long)node * outw + d] = h[((long)(n_inst + node)) * 128 + d];
  }
}

// ---------------------------------------------------------------- layernorm + lrelu + residual
__global__ __launch_bounds__(256) void finalize_kernel(float* __restrict__ h,
                                                       const float* __restrict__ s_in,
                                                       const float* __restrict__ g,
                                                       const float* __restrict__ b,
                                                       float* __restrict__ out, int n_total,
                                                       int n_inst, int lcol, int outw) {
  int wave = threadIdx.x >> 5;
  int lane = threadIdx.x & 31;
  long m = (long)blockIdx.x * 8 + wave;
  if (m >= (long)n_total) return;
  int d = lane << 2;
  const float4 s4 = *(const float4*)(s_in + m * 128 + d);
  float sum = s4.x + s4.y + s4.z + s4.w;
#pragma unroll
  for (int o = 16; o > 0; o >>= 1) sum += __shfl_xor(sum, o);
  float mu = sum * (1.0f / 128.0f);
  float dx = s4.x - mu, dy = s4.y - mu, dz = s4.z - mu, dw = s4.w - mu;
  float vs = dx * dx + dy * dy + dz * dz + dw * dw;
#pragma unroll
  for (int o = 16; o > 0; o >>= 1) vs += __shfl_xor(vs, o);
  float is = rsqrtf(vs * (1.0f / 128.0f) + 1e-5f);
  const float4 g4 = *(const float4*)(g + d);
  const float4 b4 = *(const float4*)(b + d);
  float4 h4 = *(const float4*)(h + m * 128 + d);
  float4 r;
  r.x = lrelu(dx * is * g4.x + b4.x) + h4.x;
  r.y = lrelu(dy * is * g4.y + b4.y) + h4.y;
  r.z = lrelu(dz * is * g4.z + b4.z) + h4.z;
  r.w = lrelu(dw * is * g4.w + b4.w) + h4.w;
  *(float4*)(h + m * 128 + d) = r;
  if (m >= (long)n_inst) {
    *(float4*)(out + (m - n_inst) * outw + lcol * 128 + d) = r;
  }
}

// ---------------------------------------------------------------- launch
extern "C" void kernel_launch(void* const* d_in, const int* in_sizes, int n_in, void* d_out,
                              int out_size, void* d_ws, size_t ws_size, hipStream_t stream) {
  const float* x       = (const float*)d_in[0];
  const float* x_net   = (const float*)d_in[1];
  const float* enc_W1  = (const float*)d_in[2];
  const float* enc_b1  = (const float*)d_in[3];
  const float* enc_W2  = (const float*)d_in[4];
  const float* enc_b2  = (const float*)d_in[5];
  const float* encn_W1 = (const float*)d_in[6];
  const float* encn_b1 = (const float*)d_in[7];
  const float* encn_W2 = (const float*)d_in[8];
  const float* encn_b2 = (const float*)d_in[9];
  const float* conv_W  = (const float*)d_in[10];
  const float* conv_b  = (const float*)d_in[11];
  const float* conv_rt = (const float*)d_in[12];
  const float* rconv_W = (const float*)d_in[13];
  const float* rconv_b = (const float*)d_in[14];
  const float* rconv_rt= (const float*)d_in[15];
  const float* ln_g    = (const float*)d_in[16];
  const float* ln_b    = (const float*)d_in[17];
  const int*   srcp    = (const int*)d_in[18];
  const int*   dstp    = (const int*)d_in[19];

  const int n_inst  = in_sizes[0] / 11;
  const int n_net   = in_sizes[1] / 3;
  const int n_total = n_inst + n_net;
  const int E       = in_sizes[18];
  const int outw    = out_size / n_net;  // (L+1)*128 = 512
  float* outp = (float*)d_out;

  // workspace layout (bytes)
  char* ws = (char*)d_ws;
  const size_t ND4 = (size_t)n_total * 128 * 4;
  float* h    = (float*)(ws);             // [N,128]
  float* msg  = (float*)(ws + ND4);       // [N,128]   (aliased by h1_inst)
  float* accb = (float*)(ws + 2 * ND4);   // [N,128]   (aliased by h1_inst tail)
  float* h1i  = msg;                      // [n_inst,256] spans msg+acc regions
  float* h1n  = (float*)(ws + 3 * ND4);   // [n_net,128]
  char* p2 = ws + 3 * ND4 + (size_t)n_net * 128 * 4;
  unsigned* degu_in  = (unsigned*)(p2);
  unsigned* degu_out = degu_in + n_total;
  float* dinv_in  = (float*)(degu_out + n_total);
  float* rdeg_in  = dinv_in + n_total;
  float* dinv_out = rdeg_in + n_total;
  float* rdeg_out = dinv_out + n_total;

  // degrees
  zero_u32_kernel<<<(2 * n_total + 255) / 256, 256, 0, stream>>>(degu_in, 2 * n_total);
  edge_deg_kernel<<<(E + 255) / 256, 256, 0, stream>>>(srcp, dstp, degu_in, degu_out, E);
  deg_fin_kernel<<<(n_total + 255) / 256, 256, 0, stream>>>(degu_in, degu_out, dinv_in, rdeg_in,
                                                            dinv_out, rdeg_out, n_total);

  // encoders
  enc1_inst_kernel<<<n_inst, 256, 0, stream>>>(x, enc_W1, enc_b1, h1i, n_inst);
  enc1_net_kernel<<<n_net, 128, 0, stream>>>(x_net, encn_W1, encn_b1, h1n, n_net);

  const size_t sh256 = 128 * (256 + 2) * sizeof(float);
  const size_t sh128 = 128 * (128 + 2) * sizeof(float);
  wmma_gemm_kernel<<<(n_inst + 255) / 256, 256, sh256, stream>>>(
      h1i, enc_W2, enc_b2, n_inst, 256, 0, h, nullptr, nullptr, nullptr, nullptr, nullptr);
  wmma_gemm_kernel<<<(n_net + 255) / 256, 256, sh128, stream>>>(
      h1n, encn_W2, encn_b2, n_net, 128, 0, h + (size_t)n_inst * 128, nullptr, nullptr, nullptr,
      nullptr, nullptr);

  copy_net_h0_kernel<<<(n_net * 128 + 255) / 256, 256, 0, stream>>>(h, outp, n_net, n_inst, outw);

  const int gemmN_grid = (n_total + 255) / 256;
  const int scat_grid = (int)(((long)E * 32 + 255) / 256);
  for (int l = 0; l < 3; l++) {
    wmma_gemm_kernel<<<gemmN_grid, 256, sh128, stream>>>(
        h, conv_W + (size_t)l * 128 * 128, conv_b + l * 128, n_total, 128, 1, nullptr, msg, accb,
        conv_rt + l * 128, dinv_in, rdeg_in);
    scatter_kernel<<<scat_grid, 256, 0, stream>>>(srcp, dstp, dinv_in, msg, accb, E);
    wmma_gemm_kernel<<<gemmN_grid, 256, sh128, stream>>>(
        h, rconv_W + (size_t)l * 128 * 128, rconv_b + l * 128, n_total, 128, 2, nullptr, msg, accb,
        rconv_rt + l * 128, dinv_out, rdeg_out);
    scatter_kernel<<<scat_grid, 256, 0, stream>>>(dstp, srcp, dinv_out, msg, accb, E);
    finalize_kernel<<<(n_total + 7) / 8, 256, 0, stream>>>(h, accb, ln_g + l * 128, ln_b + l * 128,
                                                           outp, n_total, n_inst, l + 1, outw);
  }
}